// Summarizer_32435593019624
// MI455X (gfx1250) — compile-verified
//
#include <hip/hip_runtime.h>

// CDNA5 / gfx1250 overlap-add (scatter -> gather reformulation).
// out[b,p] = sum_i [p >= idx[b,i]*256] * x[b, i, p - idx[b,i]*256]
// 16-event reduction runs on the matrix pipe: V_WMMA_F32_16X16X4_F32 with an
// all-ones A-matrix => D[m,n] = sum_k B[k,n] (f32-exact accumulate).

typedef __attribute__((ext_vector_type(2))) float v2f;
typedef __attribute__((ext_vector_type(8))) float v8f;

#define NS   32768   // samples per event
#define NE   16      // events per batch
#define HOP  256     // start = idx * HOP

__global__ __launch_bounds__(256) void overlap_add_wmma(
    const float* __restrict__ x,      // (B, NE, NS)
    const int*   __restrict__ indices,// (B, NE)
    float*       __restrict__ out)    // (B, NS)
{
    const int b    = blockIdx.y;
    const int lane = threadIdx.x & 31;
    const int wave = threadIdx.x >> 5;
    const int n    = lane & 15;   // WMMA column index (N = lane%16)
    const int half = lane >> 4;   // which 2 of the 4 k-slots this lane feeds

    const float* xb = x + (size_t)b * NE * NS;  // uniform base (SGPR pair)

    // Event starts: uniform addresses -> scalar loads into SGPRs.
    int start[NE];
#pragma unroll
    for (int e = 0; e < NE; ++e) start[e] = indices[b * NE + e] * HOP;

    // Hoisted per-lane constants. Chunk c covers events 4c..4c+3; this lane
    // supplies events eA=4c+2*half and eB=eA+1. One cndmask per value (two
    // SGPR choices), not a 16-deep select chain.
    int baseA[4], baseB[4], cA0[4], cB0[4], cA1[4], cB1[4];
#pragma unroll
    for (int c = 0; c < 4; ++c) {
        const int sA = half ? start[4 * c + 2] : start[4 * c];
        const int sB = half ? start[4 * c + 3] : start[4 * c + 1];
        const int eA = 4 * c + 2 * half;
        baseA[c] = eA * NS;               // event stream base (also clamp floor)
        baseB[c] = (eA + 1) * NS;
        cA0[c] = baseA[c] + n - sA;       // chain0: positions tile + n
        cB0[c] = baseB[c] + n - sB;
        cA1[c] = baseA[c] + n + 16 - sA;  // chain1: positions tile + 16 + n
        cB1[c] = baseB[c] + n + 16 - sB;
    }

    v2f ones; ones.x = 1.0f; ones.y = 1.0f;  // A = ones -> column sums

    for (int tile = blockIdx.x * 256 + wave * 32; tile < NS;
         tile += gridDim.x * 256) {
        v8f acc0 = {}; v8f acc1 = {};

#pragma unroll
        for (int c = 0; c < 4; ++c) {
            const int tA0 = tile + cA0[c];
            const int tB0 = tile + cB0[c];
            const int tA1 = tile + cA1[c];
            const int tB1 = tile + cB1[c];

            // Branchless gather: clamp index into this event's own stream
            // (always in-bounds), zero the value for pre-window lanes.
            const float vA0 = xb[tA0 < baseA[c] ? baseA[c] : tA0];
            const float vB0 = xb[tB0 < baseB[c] ? baseB[c] : tB0];
            const float vA1 = xb[tA1 < baseA[c] ? baseA[c] : tA1];
            const float vB1 = xb[tB1 < baseB[c] ? baseB[c] : tB1];

            v2f b0, b1;
            b0.x = (tA0 >= baseA[c]) ? vA0 : 0.0f;
            b0.y = (tB0 >= baseB[c]) ? vB0 : 0.0f;
            b1.x = (tA1 >= baseA[c]) ? vA1 : 0.0f;
            b1.y = (tB1 >= baseB[c]) ? vB1 : 0.0f;

            // D = ones(16x4) * B(4x16) + C : 4 events accumulated per issue;
            // two independent chains keep the matrix pipe busy.
            acc0 = __builtin_amdgcn_wmma_f32_16x16x4_f32(
                false, ones, false, b0, (short)0, acc0, false, false);
            acc1 = __builtin_amdgcn_wmma_f32_16x16x4_f32(
                false, ones, false, b1, (short)0, acc1, false, false);
        }

        // All D rows are identical, so acc[0] holds the column sum for
        // n = lane%16 on every lane. Lanes 0-15 emit tile0, lanes 16-31 emit
        // tile1: one fully coalesced 128B store per wave.
        const float r = (half == 0) ? acc0[0] : acc1[0];
        out[(size_t)b * NS + tile + lane] = r;
    }
}

extern "C" void kernel_launch(void* const* d_in, const int* in_sizes, int n_in,
                              void* d_out, int out_size, void* d_ws, size_t ws_size,
                              hipStream_t stream) {
    const float* x   = (const float*)d_in[0];
    const int*   idx = (const int*)d_in[1];
    float*       out = (float*)d_out;

    const int batches = in_sizes[1] / NE;   // (B, NE) index array -> B

    dim3 grid(32, batches);                 // 32 tile-sweeps x B batches
    overlap_add_wmma<<<grid, 256, 0, stream>>>(x, idx, out);
}